// ChebNetGCN_34531537059965
// MI455X (gfx1250) — compile-verified
//
#include <hip/hip_runtime.h>
#include <hip/hip_bf16.h>

#define N_NODES 100000
#define N_EDGES 1600000
#define FEAT 128
#define KCH 384   // 3 * FEAT (Chebyshev concat width)

typedef float v2f __attribute__((ext_vector_type(2)));
typedef float v8f __attribute__((ext_vector_type(8)));

// Guaranteed HW fp32 atomic add at device scope (L2 atomic units), non-returning
// (STOREcnt-tracked, no wait needed by the issuing wave).
__device__ __forceinline__ void atomic_add_f32(float* p, float v) {
    asm volatile("global_atomic_add_f32 %0, %1, off scope:SCOPE_DEV"
                 :: "v"(p), "v"(v) : "memory");
}

__global__ void __launch_bounds__(256) zero_kernel(float4* __restrict__ p) {
    p[(size_t)blockIdx.x * 256 + threadIdx.x] = make_float4(0.f, 0.f, 0.f, 0.f);
}

// dst = -src  (init for fused T2 = 2*spmm(T1) - T0)
__global__ void __launch_bounds__(256) negcopy_kernel(const float4* __restrict__ src,
                                                      float4* __restrict__ dst) {
    size_t i = (size_t)blockIdx.x * 256 + threadIdx.x;
    float4 v = src[i];
    dst[i] = make_float4(-v.x, -v.y, -v.z, -v.w);
}

// Pack W (ncols x 384, row-major) into per-lane WMMA B fragments.
// Fragment f in [0,192): s = f>>6 (slab), r = f&63, g = r>>1 (k-group), h = r&1 (lane half)
//   k0 = s*128 + g*4 + h*2
//   Wp[f*ncols + n] = { W[n][k0], W[n][k0+1] }
// GEMM B load for (s, g, half, n) is then ONE contiguous b64 at
//   Wp + ((s*64 + half) + 2*g)*ncols + n
__global__ void __launch_bounds__(256) pack_weights_kernel(const float* __restrict__ W,
                                                           float2* __restrict__ Wp, int ncols) {
    int idx = blockIdx.x * 256 + threadIdx.x;   // over 192*ncols
    int f = idx / ncols;
    int n = idx - f * ncols;
    int s = f >> 6, r = f & 63, g = r >> 1, h = r & 1;
    int k0 = s * 128 + g * 4 + h * 2;
    Wp[idx] = make_float2(W[n * KCH + k0], W[n * KCH + k0 + 1]);
}

// One wave per edge: Out[row[e]] += scale*w[e] * X[col[e]]  (128 features, float4/lane)
__global__ void __launch_bounds__(256) spmm_scatter_kernel(
        const int* __restrict__ row, const int* __restrict__ col,
        const float* __restrict__ w, float scale,
        const float* __restrict__ X, float* __restrict__ Out) {
    int e = blockIdx.x * 8 + (threadIdx.x >> 5);
    int lane = threadIdx.x & 31;
    int r = row[e];
    int c = col[e];
    float wt = w[e] * scale;
    float4 v = ((const float4*)(X + (size_t)c * FEAT))[lane];
    float* o = Out + (size_t)r * FEAT + (lane << 2);
    atomic_add_f32(o + 0, wt * v.x);
    atomic_add_f32(o + 1, wt * v.y);
    atomic_add_f32(o + 2, wt * v.z);
    atomic_add_f32(o + 3, wt * v.w);
}

// out[m,n] = act( [T0|T1|T2] @ W.T + bias (+resid) ), fp32 WMMA 16x16x4.
// Each wave: 5 M-tiles (80 rows) x one 16-col N-tile; B fragment loaded once per
// k-step and reused across the 5 M-tiles. 100000 = 1250 blocks * 80 rows exact.
template <int NCOLS, bool RELU, bool RESID>
__global__ void __launch_bounds__(256) cheb_gemm_kernel(
        const float* __restrict__ T0, const float* __restrict__ T1,
        const float* __restrict__ T2, const float2* __restrict__ Wp,
        const float* __restrict__ bias, const float* __restrict__ resid,
        float* __restrict__ out) {
    constexpr int MT = 5;
    const int wave  = threadIdx.x >> 5;
    const int lane  = threadIdx.x & 31;
    const int laneM = lane & 15;
    const int half  = lane >> 4;          // K-pair {0,1} vs {2,3}
    const int m0 = blockIdx.x * (16 * MT);
    const int n0 = wave * 16;

    const float* Ts[3] = {T0, T1, T2};
    v8f acc[MT];
    #pragma unroll
    for (int t = 0; t < MT; ++t) acc[t] = v8f{0.f,0.f,0.f,0.f,0.f,0.f,0.f,0.f};

    #pragma unroll
    for (int s = 0; s < 3; ++s) {
        const float* Abase[MT];
        #pragma unroll
        for (int t = 0; t < MT; ++t)
            Abase[t] = Ts[s] + (size_t)(m0 + t * 16 + laneM) * FEAT + 2 * half;
        const float2* Bbase = Wp + ((size_t)(s * 64 + half)) * NCOLS + n0 + laneM;
        #pragma unroll 4
        for (int g = 0; g < 32; ++g) {
            v2f b = *(const v2f*)(Bbase + (size_t)(2 * g) * NCOLS);   // one b64, coalesced
            #pragma unroll
            for (int t = 0; t < MT; ++t) {
                v2f a = *(const v2f*)(Abase[t] + 4 * g);              // one b64
                acc[t] = __builtin_amdgcn_wmma_f32_16x16x4_f32(
                        false, a, false, b, (short)0, acc[t], false, false);
            }
        }
    }

    const int n = n0 + laneM;
    const float bv = bias[n];
    #pragma unroll
    for (int t = 0; t < MT; ++t) {
        #pragma unroll
        for (int r = 0; r < 8; ++r) {
            int m = m0 + t * 16 + 8 * half + r;
            float v = acc[t][r] + bv;
            if (RESID) v += resid[(size_t)m * NCOLS + n];
            if (RELU)  v = fmaxf(v, 0.f);
            out[(size_t)m * NCOLS + n] = v;
        }
    }
}

extern "C" void kernel_launch(void* const* d_in, const int* in_sizes, int n_in,
                              void* d_out, int out_size, void* d_ws, size_t ws_size,
                              hipStream_t stream) {
    const float* x     = (const float*)d_in[0];
    const int*   ei    = (const int*)d_in[1];
    const float* ew    = (const float*)d_in[2];
    const float* W_in  = (const float*)d_in[3];
    const float* b_in  = (const float*)d_in[4];
    const float* W_h1  = (const float*)d_in[5];
    const float* b_h1  = (const float*)d_in[6];
    const float* W_h2  = (const float*)d_in[7];
    const float* b_h2  = (const float*)d_in[8];
    const float* W_out = (const float*)d_in[9];
    const float* b_out = (const float*)d_in[10];
    const int* row = ei;
    const int* col = ei + N_EDGES;

    float* ws = (float*)d_ws;
    const size_t NF = (size_t)N_NODES * FEAT;
    float*  Ha    = ws;
    float*  Hb    = ws + NF;
    float*  T1    = ws + 2 * NF;
    float*  T2    = ws + 3 * NF;
    float2* WpIn  = (float2*)(ws + 4 * NF);           // 192*128 float2 each
    float2* WpH1  = WpIn + 192 * FEAT;
    float2* WpH2  = WpH1 + 192 * FEAT;
    float2* WpOut = WpH2 + 192 * FEAT;                // 192*64 float2

    dim3 b256(256);
    pack_weights_kernel<<<(192 * FEAT) / 256, b256, 0, stream>>>(W_in,  WpIn,  FEAT);
    pack_weights_kernel<<<(192 * FEAT) / 256, b256, 0, stream>>>(W_h1,  WpH1,  FEAT);
    pack_weights_kernel<<<(192 * FEAT) / 256, b256, 0, stream>>>(W_h2,  WpH2,  FEAT);
    pack_weights_kernel<<<(192 * 64)   / 256, b256, 0, stream>>>(W_out, WpOut, 64);

    const int zeroBlocks = (int)(NF / 4 / 256);   // 12500 (exact)
    const int edgeBlocks = N_EDGES / 8;           // 200000 (exact)
    const int gemmBlocks = N_NODES / 80;          // 1250 (exact, 5 M-tiles/wave)

    auto spmm_pair = [&](const float* T0in) {
        zero_kernel<<<zeroBlocks, b256, 0, stream>>>((float4*)T1);
        spmm_scatter_kernel<<<edgeBlocks, b256, 0, stream>>>(row, col, ew, 1.0f, T0in, T1);
        negcopy_kernel<<<zeroBlocks, b256, 0, stream>>>((const float4*)T0in, (float4*)T2);
        spmm_scatter_kernel<<<edgeBlocks, b256, 0, stream>>>(row, col, ew, 2.0f, T1, T2);
    };

    // Layer 1: h = relu(cheb(x))
    spmm_pair(x);
    cheb_gemm_kernel<FEAT, true, false><<<gemmBlocks, b256, 0, stream>>>(
        x, T1, T2, WpIn, b_in, nullptr, Ha);
    // Layer 2: h = relu(cheb(h) + h)
    spmm_pair(Ha);
    cheb_gemm_kernel<FEAT, true, true><<<gemmBlocks, b256, 0, stream>>>(
        Ha, T1, T2, WpH1, b_h1, Ha, Hb);
    // Layer 3: h = relu(cheb(h) + h)
    spmm_pair(Hb);
    cheb_gemm_kernel<FEAT, true, true><<<gemmBlocks, b256, 0, stream>>>(
        Hb, T1, T2, WpH2, b_h2, Hb, Ha);
    // Layer 4: out = cheb(h)  (64 cols -> 4 waves/block)
    spmm_pair(Ha);
    cheb_gemm_kernel<64, false, false><<<gemmBlocks, dim3(128), 0, stream>>>(
        Ha, T1, T2, WpOut, b_out, nullptr, (float*)d_out);

    (void)in_sizes; (void)n_in; (void)out_size; (void)ws_size;
}